// MaskedMiniPointNet_73383811219637
// MI455X (gfx1250) — compile-verified
//
#include <hip/hip_runtime.h>
#include <math.h>

typedef float v2f __attribute__((ext_vector_type(2)));
typedef float v8f __attribute__((ext_vector_type(8)));

#define LN_EPS 1e-5f

constexpr int B_  = 32;
constexpr int N_  = 4096;
constexpr int BM  = 32;          // rows (points) per workgroup tile
constexpr int RB  = N_ / BM;     // 128 row-blocks per batch

// ---- monotone float <-> uint encoding so integer atomicMax == float max ----
__device__ __forceinline__ unsigned encf(float f) {
  unsigned u = __float_as_uint(f);
  return u ^ (unsigned)(((int)u >> 31) | 0x80000000);
}
__device__ __forceinline__ float decf(unsigned e) {
  unsigned u = (e & 0x80000000u) ? (e ^ 0x80000000u) : ~e;
  return __uint_as_float(u);
}

// ---------------------------------------------------------------- init gmax
__global__ __launch_bounds__(256) void k_init(unsigned* __restrict__ gmax_enc, int n) {
  int i = blockIdx.x * 256 + threadIdx.x;
  if (i < n) gmax_enc[i] = 0u;   // 0 encodes below every real float
}

// --------------------------------------------------- stage1: conv1+LN+conv2
// block: one (batch, 32-row) tile. 256 threads = 8 wave32.
// wave w owns BOTH row-tiles of col-tiles {2w, 2w+1}: B loaded once, used 2x.
__global__ __launch_bounds__(256) void k_stage1(
    const float* __restrict__ points, const int* __restrict__ mask,
    const float* __restrict__ w1, const float* __restrict__ g1,
    const float* __restrict__ be1,
    const float* __restrict__ w2, const float* __restrict__ b2,
    float* __restrict__ h3, unsigned* __restrict__ gmax_enc) {
  const int blk = blockIdx.x;
  const int b  = blk / RB;
  const int rb = blk % RB;
  const int n0 = rb * BM;
  const int t  = threadIdx.x;

  __shared__ float sA[BM][132];       // h2 tile (32 x 128), stride 132 (%64==4)
  __shared__ float sP[BM][3];
  __shared__ float sW1[128 * 3];
  __shared__ float sG1[128], sB1[128];

  if (t < 128) { sG1[t] = g1[t]; sB1[t] = be1[t]; }
  for (int i = t; i < 384; i += 256) sW1[i] = w1[i];
  if (t < BM * 3) sP[t / 3][t % 3] = points[(size_t)b * N_ * 3 + (size_t)n0 * 3 + t];
  __syncthreads();

  // conv1 + LN1 + mask + relu -> sA. thread t: row t>>3, 16 channels.
  {
    const int row = t >> 3;
    const int c0  = (t & 7) * 16;
    const float p0 = sP[row][0], p1 = sP[row][1], p2 = sP[row][2];
    float vals[16];
    float s = 0.f, ss = 0.f;
#pragma unroll
    for (int i = 0; i < 16; ++i) {
      const int c = c0 + i;
      float v = p0 * sW1[c * 3 + 0] + p1 * sW1[c * 3 + 1] + p2 * sW1[c * 3 + 2];
      vals[i] = v; s += v; ss += v * v;
    }
    for (int m = 1; m < 8; m <<= 1) {      // 8 threads own one row
      s  += __shfl_xor(s,  m, 32);
      ss += __shfl_xor(ss, m, 32);
    }
    const float mu  = s * (1.f / 128.f);
    const float var = ss * (1.f / 128.f) - mu * mu;
    const float rs  = rsqrtf(var + LN_EPS);
    const float vis = (mask[b * N_ + n0 + row] != 0) ? 1.f : 0.f;
#pragma unroll
    for (int i = 0; i < 16; ++i) {
      const int c = c0 + i;
      float v = ((vals[i] - mu) * rs * sG1[c] + sB1[c]) * vis;
      sA[row][c] = fmaxf(v, 0.f);
    }
  }
  __syncthreads();

  // GEMM: (32x128) x (128x256) via V_WMMA_F32_16X16X4_F32
  const int w  = t >> 5, l = t & 31;
  const int ln = l & 15;
  const int ksel = 2 * (l >> 4);

  v8f acc[2][2] = {};                       // [row-tile][col-tile]
  for (int k = 0; k < 128; k += 4) {
    v2f a0 = *(const v2f*)&sA[ln][k + ksel];
    v2f a1 = *(const v2f*)&sA[16 + ln][k + ksel];
#pragma unroll
    for (int i = 0; i < 2; ++i) {
      const int ncol = (w * 2 + i) * 16 + ln;
      v2f bf = *(const v2f*)&w2[(size_t)ncol * 128 + k + ksel];
      acc[0][i] = __builtin_amdgcn_wmma_f32_16x16x4_f32(
          false, a0, false, bf, (short)0, acc[0][i], false, false);
      acc[1][i] = __builtin_amdgcn_wmma_f32_16x16x4_f32(
          false, a1, false, bf, (short)0, acc[1][i], false, false);
    }
  }

  // epilogue: +b2, store h3, column-max (both row-tiles in-register)
  float* h3p = h3 + ((size_t)b * N_ + n0) * 256;
#pragma unroll
  for (int i = 0; i < 2; ++i) {
    const int ncol = (w * 2 + i) * 16 + ln;
    const float bias = b2[ncol];
    float cmax = -INFINITY;
#pragma unroll
    for (int rt = 0; rt < 2; ++rt) {
#pragma unroll
      for (int r = 0; r < 8; ++r) {
        const int m = rt * 16 + 8 * (l >> 4) + r;
        float v = acc[rt][i][r] + bias;
        h3p[(size_t)m * 256 + ncol] = v;
        cmax = fmaxf(cmax, v);
      }
    }
    cmax = fmaxf(cmax, __shfl_xor(cmax, 16, 32));   // merge row-half lanes (rows 0-7 / 8-15)
    if (l < 16) atomicMax(&gmax_enc[b * 256 + ncol], encf(cmax));
  }
}

// ------------------------------- cvec[b][p] = gmax[b] . w3[p][0:256]
__global__ __launch_bounds__(512) void k_cvec(
    const unsigned* __restrict__ gmax_enc, const float* __restrict__ w3,
    float* __restrict__ cvec) {
  const int b = blockIdx.x;
  const int t = threadIdx.x;
  __shared__ float sg[256];
  if (t < 256) sg[t] = decf(gmax_enc[b * 256 + t]);
  __syncthreads();
  const float* wr = w3 + (size_t)t * 512;
  float s = 0.f;
  for (int o = 0; o < 256; ++o) s += sg[o] * wr[o];
  cvec[b * 512 + t] = s;
}

// ------------------- stage2: conv3(+concat const)+LN2+relu+conv4+row-max
__global__ __launch_bounds__(256) void k_stage2(
    const float* __restrict__ h3, const int* __restrict__ mask,
    const float* __restrict__ w3, const float* __restrict__ g2,
    const float* __restrict__ be2,
    const float* __restrict__ w4, const float* __restrict__ b4,
    const float* __restrict__ cvec, float* __restrict__ partial) {
  const int blk = blockIdx.x;
  const int b  = blk / RB;
  const int rb = blk % RB;
  const int n0 = rb * BM;
  const int t  = threadIdx.x;

  __shared__ float smem[BM * 516];                 // 66 KB, aliased buffers
  float (*sA1)[260] = (float(*)[260])smem;         // h3 tile 32x256 (phase 1)
  float (*sA2)[516] = (float(*)[516])smem;         // h6 tile 32x512 (phase 2)

  // load h3 tile -> sA1 (float4)
  {
    const float* src = h3 + ((size_t)b * N_ + n0) * 256;
    for (int i = t * 4; i < BM * 256; i += 256 * 4) {
      const int row = i >> 8, col = i & 255;
      float4 v = *(const float4*)&src[(size_t)row * 256 + col];
      *(float4*)&sA1[row][col] = v;
    }
  }
  __syncthreads();

  const int w  = t >> 5, l = t & 31;
  const int ln = l & 15;
  const int ksel = 2 * (l >> 4);

  // GEMM1: (32x256) x (256x512), B[k][p] = w3[p][256+k]
  // wave w owns both row-tiles of col-tiles {4w .. 4w+3}
  v8f acc1[2][4] = {};
  for (int k = 0; k < 256; k += 4) {
    v2f a0 = *(const v2f*)&sA1[ln][k + ksel];
    v2f a1 = *(const v2f*)&sA1[16 + ln][k + ksel];
#pragma unroll
    for (int i = 0; i < 4; ++i) {
      const int p = (w * 4 + i) * 16 + ln;
      v2f bf = *(const v2f*)&w3[(size_t)p * 512 + 256 + k + ksel];
      acc1[0][i] = __builtin_amdgcn_wmma_f32_16x16x4_f32(
          false, a0, false, bf, (short)0, acc1[0][i], false, false);
      acc1[1][i] = __builtin_amdgcn_wmma_f32_16x16x4_f32(
          false, a1, false, bf, (short)0, acc1[1][i], false, false);
    }
  }
  __syncthreads();                                  // sA1 dead; reuse as sA2

  // h4 = acc1 + cvec -> sA2
#pragma unroll
  for (int i = 0; i < 4; ++i) {
    const int p = (w * 4 + i) * 16 + ln;
    const float cv = cvec[b * 512 + p];
#pragma unroll
    for (int rt = 0; rt < 2; ++rt) {
#pragma unroll
      for (int r = 0; r < 8; ++r) {
        const int m = rt * 16 + 8 * (l >> 4) + r;
        sA2[m][p] = acc1[rt][i][r] + cv;
      }
    }
  }
  __syncthreads();

  // LN2 + mask + relu in place (two passes, low register use)
  {
    const int row = t >> 3;
    const int c0  = (t & 7) * 64;
    float s = 0.f, ss = 0.f;
    for (int i = 0; i < 64; ++i) {
      float v = sA2[row][c0 + i];
      s += v; ss += v * v;
    }
    for (int m = 1; m < 8; m <<= 1) {
      s  += __shfl_xor(s,  m, 32);
      ss += __shfl_xor(ss, m, 32);
    }
    const float mu  = s * (1.f / 512.f);
    const float var = ss * (1.f / 512.f) - mu * mu;
    const float rs  = rsqrtf(var + LN_EPS);
    const float vis = (mask[b * N_ + n0 + row] != 0) ? 1.f : 0.f;
    for (int i = 0; i < 64; ++i) {
      const int c = c0 + i;
      float v = ((sA2[row][c] - mu) * rs * g2[c] + be2[c]) * vis;
      sA2[row][c] = fmaxf(v, 0.f);
    }
  }
  __syncthreads();

  // GEMM2: (32x512) x (512x1024) in two column halves, B[k][f] = w4[f][k]
  // per half: wave w owns both row-tiles of col-tiles {h*32 + 4w .. +3}
  for (int h = 0; h < 2; ++h) {
    v8f acc2[2][4] = {};
    for (int k = 0; k < 512; k += 4) {
      v2f a0 = *(const v2f*)&sA2[ln][k + ksel];
      v2f a1 = *(const v2f*)&sA2[16 + ln][k + ksel];
#pragma unroll
      for (int i = 0; i < 4; ++i) {
        const int f = (h * 32 + w * 4 + i) * 16 + ln;
        v2f bf = *(const v2f*)&w4[(size_t)f * 512 + k + ksel];
        acc2[0][i] = __builtin_amdgcn_wmma_f32_16x16x4_f32(
            false, a0, false, bf, (short)0, acc2[0][i], false, false);
        acc2[1][i] = __builtin_amdgcn_wmma_f32_16x16x4_f32(
            false, a1, false, bf, (short)0, acc2[1][i], false, false);
      }
    }
    // epilogue: +b4, 32-row column max fully in-register, write partial
#pragma unroll
    for (int i = 0; i < 4; ++i) {
      const int f = (h * 32 + w * 4 + i) * 16 + ln;
      const float bias = b4[f];
      float cmax = -INFINITY;
#pragma unroll
      for (int rt = 0; rt < 2; ++rt)
#pragma unroll
        for (int r = 0; r < 8; ++r) cmax = fmaxf(cmax, acc2[rt][i][r] + bias);
      cmax = fmaxf(cmax, __shfl_xor(cmax, 16, 32));
      if (l < 16)
        partial[(size_t)(b * RB + rb) * 1024 + f] = cmax;
    }
  }
}

// ----------------------------------- final: max over RB partials per (b,f)
__global__ __launch_bounds__(256) void k_final(const float* __restrict__ partial,
                                               float* __restrict__ out) {
  const int b = blockIdx.x, t = threadIdx.x;
  for (int j = 0; j < 4; ++j) {
    const int f = t + j * 256;
    const float* p = partial + (size_t)b * RB * 1024 + f;
    float m = -INFINITY;
    for (int r = 0; r < RB; ++r) m = fmaxf(m, p[(size_t)r * 1024]);
    out[b * 1024 + f] = m;
  }
}

// ---------------------------------------------------------------- launcher
extern "C" void kernel_launch(void* const* d_in, const int* in_sizes, int n_in,
                              void* d_out, int out_size, void* d_ws, size_t ws_size,
                              hipStream_t stream) {
  const float* points = (const float*)d_in[0];
  const int*   mask   = (const int*)d_in[1];
  const float* w1  = (const float*)d_in[2];
  const float* g1  = (const float*)d_in[3];
  const float* be1 = (const float*)d_in[4];
  const float* w2  = (const float*)d_in[5];
  const float* b2  = (const float*)d_in[6];
  const float* w3  = (const float*)d_in[7];
  const float* g2  = (const float*)d_in[8];
  const float* be2 = (const float*)d_in[9];
  const float* w4  = (const float*)d_in[10];
  const float* b4  = (const float*)d_in[11];
  float* out = (float*)d_out;

  // workspace layout (floats)
  float* ws = (float*)d_ws;
  float*    h3       = ws;                                       // 32*4096*256
  unsigned* gmax_enc = (unsigned*)(ws + (size_t)B_ * N_ * 256);  // 32*256
  float*    cvec     = ws + (size_t)B_ * N_ * 256 + B_ * 256;    // 32*512
  float*    partial  = cvec + B_ * 512;                          // 32*128*1024

  const int nblk = B_ * RB;   // 4096

  k_init<<<(B_ * 256 + 255) / 256, 256, 0, stream>>>(gmax_enc, B_ * 256);
  k_stage1<<<nblk, 256, 0, stream>>>(points, mask, w1, g1, be1, w2, b2,
                                     h3, gmax_enc);
  k_cvec<<<B_, 512, 0, stream>>>(gmax_enc, w3, cvec);
  k_stage2<<<nblk, 256, 0, stream>>>(h3, mask, w3, g2, be2, w4, b4,
                                     cvec, partial);
  k_final<<<B_, 256, 0, stream>>>(partial, out);
}